// SnakeConv_75960791597260
// MI455X (gfx1250) — compile-verified
//
#include <hip/hip_runtime.h>

typedef __attribute__((ext_vector_type(2))) float v2f;
typedef __attribute__((ext_vector_type(8))) float v8f;

__device__ __forceinline__ v8f wmma_f32(v2f a, v2f b, v8f c) {
    // D = A(16x4) * B(4x16) + C, full f32. 8-arg VOP3P form.
    return __builtin_amdgcn_wmma_f32_16x16x4_f32(false, a, false, b, (short)0, c, false, false);
}

// ---- fragment loaders (layouts per CDNA5 ISA §7.12.2) ----
// A 16x4 f32: lane L holds row M = L%16; VGPR v holds K = 4*kc + 2*(L/16) + v
// Weight layout: W[(o*CIN + c)*TAPS + tap]
template <int CIN, int TAPS, int OMAX>
__device__ __forceinline__ v2f load_afrag(const float* __restrict__ W,
                                          int obase, int kc, int tap, int lane) {
    int o = obase + (lane & 15);
    int c = kc * 4 + ((lane >> 4) << 1);
    v2f a;
    if (o < OMAX) {
        const float* p = W + (o * CIN + c) * TAPS + tap;
        a.x = p[0];
        a.y = p[TAPS];
    } else {
        a.x = 0.f; a.y = 0.f;
    }
    return a;
}

// B 4x16 f32: lane L holds col N = L%16; VGPR v holds K = 4*kc + 2*(L/16) + v
__device__ __forceinline__ v2f load_bfrag_direct(const float* __restrict__ src, int kc,
                                                 int lane, int p) {
    int c = kc * 4 + ((lane >> 4) << 1);
    v2f b;
    b.x = src[c * 64 + p];
    b.y = src[(c + 1) * 64 + p];
    return b;
}

__device__ __forceinline__ v2f load_bfrag_shift(const float* __restrict__ src, int kc,
                                                int lane, int pp /* -1 => zero pad */) {
    int c = kc * 4 + ((lane >> 4) << 1);
    v2f b;
    if (pp >= 0) {
        b.x = src[c * 64 + pp];
        b.y = src[(c + 1) * 64 + pp];
    } else {
        b.x = 0.f; b.y = 0.f;
    }
    return b;
}

__device__ __forceinline__ v2f load_bfrag_lerp(const float* __restrict__ src, int kc, int lane,
                                               int i0, int i1, float wy) {
    int c = kc * 4 + ((lane >> 4) << 1);
    float om = 1.f - wy;
    v2f b;
    b.x = src[c * 64 + i0] * om + src[c * 64 + i1] * wy;
    b.y = src[(c + 1) * 64 + i0] * om + src[(c + 1) * 64 + i1] * wy;
    return b;
}

// ---- dense 64x64x64 GEMM (projections): dst[c][k] = sum_q W[c*64+q]*src[q][k] + bias[c]
__device__ __forceinline__ void proj_gemm(const float* __restrict__ W, const float* __restrict__ bias,
                                          const float* __restrict__ src,
                                          float* dstL, float* dstG, int wave, int lane) {
    int mo = (wave & 3) * 16;
    int p0 = (wave >> 2) * 32 + (lane & 15);
    int p1 = p0 + 16;
    v8f acc0 = {}; v8f acc1 = {};
    #pragma unroll 4
    for (int kc = 0; kc < 16; ++kc) {
        v2f a  = load_afrag<64, 1, 64>(W, mo, kc, 0, lane);
        v2f b0 = load_bfrag_direct(src, kc, lane, p0);
        acc0 = wmma_f32(a, b0, acc0);
        v2f b1 = load_bfrag_direct(src, kc, lane, p1);
        acc1 = wmma_f32(a, b1, acc1);
    }
    #pragma unroll
    for (int i = 0; i < 8; ++i) {
        int row = mo + i + ((lane >> 4) << 3);   // C/D layout: VGPR i -> M=i (lanes 0-15), M=8+i
        float v0 = acc0[i] + bias[row];
        float v1 = acc1[i] + bias[row];
        if (dstL) { dstL[row * 64 + p0] = v0; dstL[row * 64 + p1] = v1; }
        if (dstG) { dstG[row * 64 + p0] = v0; dstG[row * 64 + p1] = v1; }
    }
}

// ---- 3x3 SAME conv as 9-tap implicit GEMM. OMAX=64: 2 tiles/wave; OMAX=18: 1 tile/wave.
template <int CIN, int OMAX>
__device__ __forceinline__ void conv3x3_gemm(const float* __restrict__ W, const float* __restrict__ bias,
                                             const float* __restrict__ src, float* dst,
                                             int wave, int lane) {
    constexpr int KC = CIN / 4;
    constexpr int NT = (OMAX > 18) ? 2 : 1;
    int mo, p0;
    if (OMAX > 18) { mo = (wave & 3) * 16; p0 = (wave >> 2) * 32 + (lane & 15); }
    else           { mo = (wave & 1) * 16; p0 = (wave >> 1) * 16 + (lane & 15); }
    int p1 = p0 + 16;
    int y0 = p0 >> 3, h0 = p0 & 7;
    int y1 = p1 >> 3, h1 = p1 & 7;
    v8f acc0 = {}; v8f acc1 = {};
    #pragma unroll
    for (int tap = 0; tap < 9; ++tap) {
        int dy = tap / 3 - 1, dx = tap % 3 - 1;
        int yy0 = y0 + dy, xx0 = h0 + dx;
        int pp0 = (yy0 >= 0 && yy0 < 8 && xx0 >= 0 && xx0 < 8) ? yy0 * 8 + xx0 : -1;
        int yy1 = y1 + dy, xx1 = h1 + dx;
        int pp1 = (yy1 >= 0 && yy1 < 8 && xx1 >= 0 && xx1 < 8) ? yy1 * 8 + xx1 : -1;
        #pragma unroll 4
        for (int kc = 0; kc < KC; ++kc) {
            v2f a  = load_afrag<CIN, 9, OMAX>(W, mo, kc, tap, lane);
            v2f b0 = load_bfrag_shift(src, kc, lane, pp0);
            acc0 = wmma_f32(a, b0, acc0);
            if (NT == 2) {
                v2f b1 = load_bfrag_shift(src, kc, lane, pp1);
                acc1 = wmma_f32(a, b1, acc1);
            }
        }
    }
    #pragma unroll
    for (int i = 0; i < 8; ++i) {
        int row = mo + i + ((lane >> 4) << 3);
        if (row < OMAX) {
            dst[row * 64 + p0] = acc0[i] + bias[row];
            if (NT == 2) dst[row * 64 + p1] = acc1[i] + bias[row];
        }
    }
}

// ---- deformable 9-tap conv: B elements are lerped samples of src at precomputed indices
__device__ __forceinline__ void dsconv_gemm(const float* __restrict__ W, const float* __restrict__ bias,
                                            const float* __restrict__ src,
                                            const int* __restrict__ i0a, const int* __restrict__ i1a,
                                            const float* __restrict__ wya,
                                            float* dst, int wave, int lane) {
    int mo = (wave & 3) * 16;
    int p0 = (wave >> 2) * 32 + (lane & 15);
    int p1 = p0 + 16;
    v8f acc0 = {}; v8f acc1 = {};
    #pragma unroll
    for (int j = 0; j < 9; ++j) {
        int   i00 = i0a[j * 64 + p0], i01 = i1a[j * 64 + p0];
        float w0  = wya[j * 64 + p0];
        int   i10 = i0a[j * 64 + p1], i11 = i1a[j * 64 + p1];
        float w1  = wya[j * 64 + p1];
        #pragma unroll 4
        for (int kc = 0; kc < 16; ++kc) {
            v2f a  = load_afrag<64, 9, 64>(W, mo, kc, j, lane);
            v2f b0 = load_bfrag_lerp(src, kc, lane, i00, i01, w0);
            acc0 = wmma_f32(a, b0, acc0);
            v2f b1 = load_bfrag_lerp(src, kc, lane, i10, i11, w1);
            acc1 = wmma_f32(a, b1, acc1);
        }
    }
    #pragma unroll
    for (int i = 0; i < 8; ++i) {
        int row = mo + i + ((lane >> 4) << 3);
        dst[row * 64 + p0] = acc0[i] + bias[row];
        dst[row * 64 + p1] = acc1[i] + bias[row];
    }
}

// ---- GroupNorm (+ReLU or tanh) over buf[Cc][64], groups G
template <int ACT>  // 0 = relu, 1 = tanh
__device__ __forceinline__ void groupnorm(const float* __restrict__ buf,
                                          const float* __restrict__ gamma, const float* __restrict__ beta,
                                          int Cc, int G, float* dst,
                                          float* red, float* gpar, int tid) {
    int gs = Cc / G;
    int epg = gs * 64;       // elements per group
    int chunk = epg / 16;    // per-thread chunk (16 threads per group)
    int nthreads = G * 16;
    if (tid < nthreads) {
        int g = tid >> 4, idx = tid & 15;
        const float* base = buf + g * gs * 64;
        float s = 0.f, ss = 0.f;
        for (int e = idx * chunk; e < idx * chunk + chunk; ++e) {
            float v = base[e];
            s += v; ss += v * v;
        }
        red[tid] = s; red[256 + tid] = ss;
    }
    __syncthreads();
    if (tid < G) {
        float s = 0.f, ss = 0.f;
        for (int k = 0; k < 16; ++k) { s += red[tid * 16 + k]; ss += red[256 + tid * 16 + k]; }
        float mu  = s / (float)epg;
        float var = ss / (float)epg - mu * mu;
        gpar[tid * 2]     = rsqrtf(var + 1e-5f);
        gpar[tid * 2 + 1] = mu;
    }
    __syncthreads();
    if (tid < nthreads) {
        int g = tid >> 4, idx = tid & 15;
        float inv = gpar[g * 2], mu = gpar[g * 2 + 1];
        const float* base = buf + g * gs * 64;
        float* dbase = dst + g * gs * 64;
        for (int e = idx * chunk; e < idx * chunk + chunk; ++e) {
            int c = g * gs + (e >> 6);
            float v = (base[e] - mu) * inv * gamma[c] + beta[c];
            dbase[e] = (ACT == 0) ? fmaxf(v, 0.f) : tanhf(v);
        }
    }
    __syncthreads();
}

// ---- snake cumsum + sampling coordinates (one axis is always integer)
__device__ __forceinline__ void compute_coords(const float* __restrict__ off, int morph,
                                               int* i0a, int* i1a, float* wya, int tid) {
    if (tid < 64) {
        int p = tid, w = p >> 3, h = p & 7;
        const float* o = off + (morph ? 9 * 64 : 0);
        float cs[9];
        cs[4] = 0.f;
        float acc = 0.f;
        for (int j = 5; j < 9; ++j) { acc += o[j * 64 + p]; cs[j] = acc; }
        acc = 0.f;
        for (int j = 3; j >= 0; --j) { acc += o[j * 64 + p]; cs[j] = acc; }
        for (int j = 0; j < 9; ++j) {
            float frac; int lo, hi;
            if (morph == 0) {
                int xcol = min(max(h + j - 4, 0), 7);
                float y = fminf(fmaxf((float)w + cs[j], 0.f), 7.f);
                float yf = floorf(y);
                int yi = (int)yf, y2 = min(yi + 1, 7);
                frac = y - yf;
                lo = yi * 8 + xcol; hi = y2 * 8 + xcol;
            } else {
                int yrow = min(max(w + j - 4, 0), 7);
                float x = fminf(fmaxf((float)h + cs[j], 0.f), 7.f);
                float xf = floorf(x);
                int xi = (int)xf, x2 = min(xi + 1, 7);
                frac = x - xf;
                lo = yrow * 8 + xi; hi = yrow * 8 + x2;
            }
            i0a[j * 64 + p] = lo;
            i1a[j * 64 + p] = hi;
            wya[j * 64 + p] = frac;
        }
    }
}

__global__ __launch_bounds__(256) void snake_fused_kernel(
    const float* __restrict__ x,
    const float* __restrict__ w_proj1, const float* __restrict__ b_proj1,
    const float* __restrict__ w_proj2, const float* __restrict__ b_proj2,
    const float* __restrict__ c0_w, const float* __restrict__ c0_b,
    const float* __restrict__ c0_g, const float* __restrict__ c0_beta,
    const float* __restrict__ cx_off_w, const float* __restrict__ cx_off_b,
    const float* __restrict__ cx_goff_g, const float* __restrict__ cx_goff_b,
    const float* __restrict__ cx_w, const float* __restrict__ cx_b,
    const float* __restrict__ cx_g, const float* __restrict__ cx_beta,
    const float* __restrict__ cy_off_w, const float* __restrict__ cy_off_b,
    const float* __restrict__ cy_goff_g, const float* __restrict__ cy_goff_b,
    const float* __restrict__ cy_w, const float* __restrict__ cy_b,
    const float* __restrict__ cy_g, const float* __restrict__ cy_beta,
    const float* __restrict__ c1_w, const float* __restrict__ c1_b,
    const float* __restrict__ c1_g, const float* __restrict__ c1_beta,
    float* __restrict__ out) {

    __shared__ float s_xraw[64 * 64];   // raw input; later reused for final feature map
    __shared__ float s_xin[64 * 64];    // projected input (dsconv sampling source)
    __shared__ float s_cat[192 * 64];   // x1 | x2 | x3
    __shared__ float s_out[64 * 64];    // pre-GN conv output scratch
    __shared__ float s_off[18 * 64];    // tanh(GN(offset conv))
    __shared__ int   s_i0[9 * 64];
    __shared__ int   s_i1[9 * 64];
    __shared__ float s_wy[9 * 64];
    __shared__ float s_red[512];
    __shared__ float s_gn[32];

    const int tid = threadIdx.x;
    const int lane = tid & 31;
    const int wave = tid >> 5;
    const long base = (long)blockIdx.x * 4096;

    // stage in raw image (64 q-rows x 64 k-cols), coalesced
    for (int i = tid; i < 4096; i += 256) s_xraw[i] = x[base + i];
    __syncthreads();

    // proj1: xin[c][k] = sum_q W1[c,q] * xraw[q,k] + b1[c]
    proj_gemm(w_proj1, b_proj1, s_xraw, s_xin, nullptr, wave, lane);
    __syncthreads();

    // ---- branch 1: encoder conv c0 -> x1 (cat rows 0..63)
    conv3x3_gemm<64, 64>(c0_w, c0_b, s_xin, s_out, wave, lane);
    __syncthreads();
    groupnorm<0>(s_out, c0_g, c0_beta, 64, 16, s_cat, s_red, s_gn, tid);

    // ---- branch 2: dsconv morph=0 (cx) -> x2 (cat rows 64..127)
    conv3x3_gemm<64, 18>(cx_off_w, cx_off_b, s_xin, s_out, wave, lane);
    __syncthreads();
    groupnorm<1>(s_out, cx_goff_g, cx_goff_b, 18, 9, s_off, s_red, s_gn, tid);
    compute_coords(s_off, 0, s_i0, s_i1, s_wy, tid);
    __syncthreads();
    dsconv_gemm(cx_w, cx_b, s_xin, s_i0, s_i1, s_wy, s_out, wave, lane);
    __syncthreads();
    groupnorm<0>(s_out, cx_g, cx_beta, 64, 16, s_cat + 64 * 64, s_red, s_gn, tid);

    // ---- branch 3: dsconv morph=1 (cy) -> x3 (cat rows 128..191)
    conv3x3_gemm<64, 18>(cy_off_w, cy_off_b, s_xin, s_out, wave, lane);
    __syncthreads();
    groupnorm<1>(s_out, cy_goff_g, cy_goff_b, 18, 9, s_off, s_red, s_gn, tid);
    compute_coords(s_off, 1, s_i0, s_i1, s_wy, tid);
    __syncthreads();
    dsconv_gemm(cy_w, cy_b, s_xin, s_i0, s_i1, s_wy, s_out, wave, lane);
    __syncthreads();
    groupnorm<0>(s_out, cy_g, cy_beta, 64, 16, s_cat + 128 * 64, s_red, s_gn, tid);

    // ---- fuse: c1 3x3 conv over 192-channel concat -> final feature map (reuse s_xraw)
    conv3x3_gemm<192, 64>(c1_w, c1_b, s_cat, s_out, wave, lane);
    __syncthreads();
    groupnorm<0>(s_out, c1_g, c1_beta, 64, 16, s_xraw, s_red, s_gn, tid);

    // proj2 -> global output
    proj_gemm(w_proj2, b_proj2, s_xraw, nullptr, out + base, wave, lane);
}

extern "C" void kernel_launch(void* const* d_in, const int* in_sizes, int n_in,
                              void* d_out, int out_size, void* d_ws, size_t ws_size,
                              hipStream_t stream) {
    const float* x         = (const float*)d_in[0];
    const float* w_proj1   = (const float*)d_in[1];
    const float* b_proj1   = (const float*)d_in[2];
    const float* w_proj2   = (const float*)d_in[3];
    const float* b_proj2   = (const float*)d_in[4];
    const float* c0_w      = (const float*)d_in[5];
    const float* c0_b      = (const float*)d_in[6];
    const float* c0_g      = (const float*)d_in[7];
    const float* c0_beta   = (const float*)d_in[8];
    const float* cx_off_w  = (const float*)d_in[9];
    const float* cx_off_b  = (const float*)d_in[10];
    const float* cx_goff_g = (const float*)d_in[11];
    const float* cx_goff_b = (const float*)d_in[12];
    const float* cx_w      = (const float*)d_in[13];
    const float* cx_b      = (const float*)d_in[14];
    const float* cx_g      = (const float*)d_in[15];
    const float* cx_beta   = (const float*)d_in[16];
    const float* cy_off_w  = (const float*)d_in[17];
    const float* cy_off_b  = (const float*)d_in[18];
    const float* cy_goff_g = (const float*)d_in[19];
    const float* cy_goff_b = (const float*)d_in[20];
    const float* cy_w      = (const float*)d_in[21];
    const float* cy_b      = (const float*)d_in[22];
    const float* cy_g      = (const float*)d_in[23];
    const float* cy_beta   = (const float*)d_in[24];
    const float* c1_w      = (const float*)d_in[25];
    const float* c1_b      = (const float*)d_in[26];
    const float* c1_g      = (const float*)d_in[27];
    const float* c1_beta   = (const float*)d_in[28];
    float* outp = (float*)d_out;

    int nImg = in_sizes[0] / 4096;  // 2048 = 128*16
    snake_fused_kernel<<<nImg, 256, 0, stream>>>(
        x, w_proj1, b_proj1, w_proj2, b_proj2,
        c0_w, c0_b, c0_g, c0_beta,
        cx_off_w, cx_off_b, cx_goff_g, cx_goff_b, cx_w, cx_b, cx_g, cx_beta,
        cy_off_w, cy_off_b, cy_goff_g, cy_goff_b, cy_w, cy_b, cy_g, cy_beta,
        c1_w, c1_b, c1_g, c1_beta, outp);
}